// GNNILSModel_46059229282881
// MI455X (gfx1250) — compile-verified
//
#include <hip/hip_runtime.h>

// Problem constants from the reference
#define B_    8
#define V_    50
#define C_    100
#define H_    64
#define LMAX_ 12
#define NEDGE 12          // 11 real edge slots + 1 zero-mask pad (3 k-groups of 4)
#define PATHS_PER_BLOCK 2 // 256 threads = 8 waves = 2 paths x 4 h-group waves
#define DEMAND_INV (1.0f / 500.0f)

typedef __attribute__((ext_vector_type(2))) float v2f;
typedef __attribute__((ext_vector_type(8))) float v8f;

// One wave32 per (path, h-group of 16 channels): 3200 waves machine-wide for
// max gather MLP. Masked edge-mean folded into the matrix pipe:
//   out[h] = sum_e feats[e][h] * (mask[e]/count)  ==  A(16x4) x B(4x16)
// accumulated over 3 k-groups with V_WMMA_F32_16X16X4_F32 (f32-precise).
__global__ __launch_bounds__(256) void gnn_path_feat_kernel(
    const float* __restrict__ ef,    // (B,V,V,C,H) f32
    const int*   __restrict__ pn,    // (B,C,LMAX)  i32
    const int*   __restrict__ plen,  // (B,C)       i32
    const float* __restrict__ dem,   // (B,C)       f32
    float*       __restrict__ out)   // (B,C,H+1)   f32
{
    __shared__ int   s_eoff[PATHS_PER_BLOCK][NEDGE];  // per-edge element offsets
    __shared__ float s_mask[PATHS_PER_BLOCK][NEDGE];  // mask[e] * 1/count

    const int lane = threadIdx.x & 31;
    const int wv   = threadIdx.x >> 5;   // 0..7
    const int pl   = wv >> 2;            // path slot within block (0..1)
    const int hg   = wv & 3;             // h-group (16 channels each)
    const int path = blockIdx.x * PATHS_PER_BLOCK + pl;   // path = b*C + c
    const int b    = path / C_;
    const int c    = path % C_;

    // ---- per-path tables, built by the hg==0 wave (lanes 0..11) ----
    if (hg == 0 && lane < NEDGE) {
        const int e  = lane;
        const int pb = path * LMAX_;
        const int u  = pn[pb + e];
        const int vv = (e < LMAX_ - 1) ? pn[pb + e + 1] : u;  // slot 11: clamp (mask=0)
        s_eoff[pl][e] = (u * V_ + vv) * (C_ * H_);
        int ne = plen[path] - 1;
        if (ne < 0) ne = 0;
        const float inv = 1.0f / (float)(ne < 1 ? 1 : ne);
        s_mask[pl][e] = (e < ne) ? inv : 0.0f;                // n_edges <= 11 always
    }
    __syncthreads();

    const int half = lane >> 4;   // ISA layout: lanes 0-15 hold K=0,1; 16-31 hold K=2,3
    const int m    = lane & 15;   // M (for A) / N (for B,D) index within lane half

    // ---- B matrices (mask/count) + edge offsets for this lane's k-slots ----
    v2f bm[3];
    int eo0[3], eo1[3];
    #pragma unroll
    for (int kg = 0; kg < 3; ++kg) {
        const int e0 = 4 * kg + 2 * half;
        bm[kg].x = s_mask[pl][e0];
        bm[kg].y = s_mask[pl][e0 + 1];
        eo0[kg]  = s_eoff[pl][e0];
        eo1[kg]  = s_eoff[pl][e0 + 1];
    }

    const float* base = ef + (size_t)b * ((size_t)V_ * V_ * C_ * H_) + (size_t)c * H_;
    const int h = 16 * hg + m;

    // Issue all 6 A-element gathers up front (one load clause), then WMMA chain.
    v2f a[3];
    #pragma unroll
    for (int kg = 0; kg < 3; ++kg) {
        a[kg].x = base[(size_t)eo0[kg] + h];
        a[kg].y = base[(size_t)eo1[kg] + h];
    }

    v8f acc = {};
    #pragma unroll
    for (int kg = 0; kg < 3; ++kg) {
        acc = __builtin_amdgcn_wmma_f32_16x16x4_f32(
            /*neg_a=*/false, a[kg],
            /*neg_b=*/false, bm[kg],
            /*c_mod=*/(short)0, acc,
            /*reuse_a=*/false, /*reuse_b=*/false);
    }

    // D[M][N]: VGPR r, lane L -> D[r + 8*(L>>4)][L&15]; identical over N, so
    // lanes with N==0 (lanes 0 and 16) each write their 8 rows.
    const size_t outbase = (size_t)path * (H_ + 1);
    if (m == 0) {
        const int hb = 16 * hg + 8 * half;
        #pragma unroll
        for (int r = 0; r < 8; ++r)
            out[outbase + hb + r] = acc[r];
    }

    if (hg == 0 && lane == 0)
        out[outbase + H_] = dem[path] * DEMAND_INV;
}

extern "C" void kernel_launch(void* const* d_in, const int* in_sizes, int n_in,
                              void* d_out, int out_size, void* d_ws, size_t ws_size,
                              hipStream_t stream) {
    (void)in_sizes; (void)n_in; (void)out_size; (void)d_ws; (void)ws_size;
    const float* ef   = (const float*)d_in[0];  // edge_features (B,V,V,C,H)
    const int*   pn   = (const int*)  d_in[1];  // path_nodes    (B,C,LMAX)
    const int*   plen = (const int*)  d_in[2];  // path_lengths  (B,C)
    const float* dem  = (const float*)d_in[3];  // demands       (B,C)
    float*       out  = (float*)d_out;          // (B,C,H+1)

    const int n_paths = B_ * C_;                        // 800
    const int grid    = n_paths / PATHS_PER_BLOCK;      // 400 blocks x 256 threads
    gnn_path_feat_kernel<<<grid, 256, 0, stream>>>(ef, pn, plen, dem, out);
}